// ParC_operator_65180423685137
// MI455X (gfx1250) — compile-verified
//
#include <hip/hip_runtime.h>

// ParC circular depthwise conv along H as per-channel circulant GEMM:
//   Out_c[56 x 1792] = A_c[56 x 56] * X_c[56 x 1792] + colbias_c[56],
//   A_c[h,j] = w_c[(j-h) mod 56],  colbias_c[h] = (A_c * pe_c)[h] + bias_c
// The PE term is column-invariant, so it folds into a per-channel H-vector,
// leaving the x staging as a pure copy -> gfx1250 async global->LDS b128 DMA.
// f32 WMMA (V_WMMA_F32_16X16X4_F32) keeps full precision; traffic is the
// streaming minimum (x read once, out written once) -> HBM-bound at 23.3 TB/s.

namespace {
constexpr int B_ = 32, C_ = 384, H_ = 56, W_ = 56;
constexpr int NCOL = B_ * W_;     // 1792 (b,w) columns per channel
constexpr int NB   = 128;         // columns per workgroup (8 waves x 16)
constexpr int NBLK = NCOL / NB;   // 14 column blocks per channel
constexpr int LDSW = NB + 4;      // padded LDS row stride (floats, mult. of 4)
}

typedef __attribute__((ext_vector_type(2))) float v2f;
typedef __attribute__((ext_vector_type(8))) float v8f;

struct SMem {
    float xs[H_ * LDSW];  // MUST be first: async-LDS asm assumes xs at LDS offset 0
    float wc[112];        // circulant weight lookup: wc[t] = w[c, t mod 56]
    float cb[64];         // per-row column bias: (A*pe)[h] + bias[c]
};

__global__ __launch_bounds__(256)
void parc_circconv_wmma(const float* __restrict__ x,
                        const float* __restrict__ pe,
                        const float* __restrict__ weight,
                        const float* __restrict__ bias,
                        float* __restrict__ out)
{
    __shared__ SMem sm;

    const int wg  = blockIdx.x;
    const int c   = wg / NBLK;
    const int n0  = (wg - c * NBLK) * NB;
    const int tid = threadIdx.x;

    // ---- Async DMA of the 56x128 f32 tile into LDS (pure copy, b128/lane) ----
    // Any 4 consecutive cols with col%4==0 stay inside one batch b and are
    // 16B-aligned in both global and LDS (W=56 and LDSW=132 are multiples of 4).
    #pragma unroll
    for (int it = 0; it < (H_ * NB / 4) / 256; ++it) {   // 7 groups per thread
        const int i   = tid + it * 256;
        const int j   = i >> 5;                // row 0..55  (32 groups per row)
        const int col = (i & 31) * 4;          // column group start
        const int n   = n0 + col;
        const int b   = n / W_;
        const int w   = n - b * W_;
        const unsigned gofs = (unsigned)((((b * C_ + c) * H_ + j) * W_ + w) * 4);
        const unsigned lofs = (unsigned)((j * LDSW + col) * 4);
        asm volatile("global_load_async_to_lds_b128 %0, %1, %2 offset:0"
                     :: "v"(lofs), "v"(gofs), "s"(x)
                     : "memory");
    }

    // ---- While the DMA flies: circulant weight table + folded PE/bias ----
    if (tid < 112) {
        int t = tid; if (t >= H_) t -= H_;
        sm.wc[tid] = weight[c * H_ + t];
    }
    if (tid < H_) {
        float s = bias[c];
        const float* wch = weight + c * H_;
        const float* peh = pe + c * H_;
        for (int j = 0; j < H_; ++j) {
            int t = j - tid; if (t < 0) t += H_;
            s += wch[t] * peh[j];
        }
        sm.cb[tid] = s;
    }

    asm volatile("s_wait_asynccnt 0" ::: "memory");
    __syncthreads();

    // ---- 64x16 output tile per wave: 4 M-tiles, K-loop of 14 x (K=4) ----
    const int lane = tid & 31;
    const int l15  = lane & 15;
    const int half = lane >> 4;                 // 0: lanes 0-15, 1: lanes 16-31
    const int ncol = (tid >> 5) * 16 + l15;     // this wave's N-tile column

    v8f acc0 = {}, acc1 = {}, acc2 = {}, acc3 = {};

    // A-operand per ISA layout: lane M = l15, VGPR v holds K = v + 2*half.
    // A[h][k] = wc[k - h + 56]; per-lane base index for v=0, mt=0:
    const int ab0 = 2 * half + 56 - l15;

    #pragma unroll
    for (int kk = 0; kk < H_ / 4; ++kk) {
        const int krow = kk * 4 + 2 * half;     // B row for v=0 in this lane
        v2f bf;
        bf.x = sm.xs[(krow + 0) * LDSW + ncol]; // B[k][n], rows striped across lanes
        bf.y = sm.xs[(krow + 1) * LDSW + ncol];

        const int i0 = ab0 + 4 * kk;            // wc index, mt=0, v=0
        v2f a0, a1, a2, a3;
        a0.x = sm.wc[i0];      a0.y = sm.wc[i0 + 1];
        a1.x = sm.wc[i0 - 16]; a1.y = sm.wc[i0 - 15];
        a2.x = sm.wc[i0 - 32]; a2.y = sm.wc[i0 - 31];
        int i3x = i0 - 48;  if (i3x < 0) i3x = 0;   // only rows h>=56 (discarded) clamp
        int i3y = i0 - 47;  if (i3y < 0) i3y = 0;
        a3.x = sm.wc[i3x];     a3.y = sm.wc[i3y];

        acc0 = __builtin_amdgcn_wmma_f32_16x16x4_f32(false, a0, false, bf, (short)0, acc0, false, false);
        acc1 = __builtin_amdgcn_wmma_f32_16x16x4_f32(false, a1, false, bf, (short)0, acc1, false, false);
        acc2 = __builtin_amdgcn_wmma_f32_16x16x4_f32(false, a2, false, bf, (short)0, acc2, false, false);
        acc3 = __builtin_amdgcn_wmma_f32_16x16x4_f32(false, a3, false, bf, (short)0, acc3, false, false);
    }

    // ---- Epilogue: D layout -> VGPR v: lanes<16 M=v, lanes>=16 M=v+8; N=l15 ----
    const int nglob = n0 + ncol;
    const int b     = nglob / W_;
    const int w     = nglob - b * W_;
    const int base  = ((b * C_ + c) * H_) * W_ + w;

    #pragma unroll
    for (int v = 0; v < 8; ++v) {
        const int h = v + 8 * half;
        out[base + (h +  0) * W_] = acc0[v] + sm.cb[h];
        out[base + (h + 16) * W_] = acc1[v] + sm.cb[h + 16];
        out[base + (h + 32) * W_] = acc2[v] + sm.cb[h + 32];
        const int h3 = h + 48;
        if (h3 < H_) out[base + h3 * W_] = acc3[v] + sm.cb[h3];
    }
}

extern "C" void kernel_launch(void* const* d_in, const int* in_sizes, int n_in,
                              void* d_out, int out_size, void* d_ws, size_t ws_size,
                              hipStream_t stream) {
    const float* x  = (const float*)d_in[0];
    const float* pe = (const float*)d_in[1];
    const float* wt = (const float*)d_in[2];
    const float* bs = (const float*)d_in[3];
    float* outp     = (float*)d_out;

    dim3 grid(C_ * NBLK);   // 384 channels * 14 column blocks = 5376 workgroups
    dim3 block(256);        // 8 wave32
    parc_circconv_wmma<<<grid, block, 0, stream>>>(x, pe, wt, bs, outp);
}